// SageForDiffPool_51788715655369
// MI455X (gfx1250) — compile-verified
//
#include <hip/hip_runtime.h>

typedef __attribute__((ext_vector_type(2))) float v2f;
typedef __attribute__((ext_vector_type(8))) float v8f;

#define CFEAT 256

// ---------------------------------------------------------------- zero
__global__ void zero_kernel(float* __restrict__ p, long n) {
    long i = (long)blockIdx.x * blockDim.x + threadIdx.x;
    long stride = (long)gridDim.x * blockDim.x;
    for (; i < n; i += stride) p[i] = 0.0f;
}

// ---------------------------------------------------------------- scatter (segment sum + count)
// One wave per edge: 32 lanes x 8 floats = 256 features.
__global__ void scatter_kernel(const float* __restrict__ x,
                               const int* __restrict__ src,
                               const int* __restrict__ dst,
                               float* __restrict__ s,
                               float* __restrict__ cnt,
                               int E) {
    int wave = (int)((blockIdx.x * blockDim.x + threadIdx.x) >> 5);
    int lane = threadIdx.x & 31;
    if (wave >= E) return;
    int sr = src[wave];
    int ds = dst[wave];
    const float4* xs = (const float4*)(x + (long)sr * CFEAT);
    float* sd = s + (long)ds * CFEAT;
    float4 a = xs[lane];        // features [lane*4 .. lane*4+3]
    float4 b = xs[lane + 32];   // features [128+lane*4 ..]
    int base = lane * 4;
    atomicAdd(&sd[base + 0], a.x);
    atomicAdd(&sd[base + 1], a.y);
    atomicAdd(&sd[base + 2], a.z);
    atomicAdd(&sd[base + 3], a.w);
    atomicAdd(&sd[128 + base + 0], b.x);
    atomicAdd(&sd[128 + base + 1], b.y);
    atomicAdd(&sd[128 + base + 2], b.z);
    atomicAdd(&sd[128 + base + 3], b.w);
    if (lane == 0) atomicAdd(&cnt[ds], 1.0f);
}

// ---------------------------------------------------------------- fused GEMM + bias + ReLU + column stats
// h = relu( (s/max(cnt,1)) @ Wl^T + bl + xt @ Wr^T )
// One wave computes a 16-row x 64-col strip (4 WMMA 16x16 tiles), f32 WMMA.
__global__ void gemm_kernel(const float* __restrict__ s,
                            const float* __restrict__ cnt,
                            const float* __restrict__ xt,
                            const float* __restrict__ Wl,
                            const float* __restrict__ bl,
                            const float* __restrict__ Wr,
                            float* __restrict__ h,
                            float* __restrict__ colsum,
                            float* __restrict__ colsq,
                            int N, int rowTiles) {
    int wavesPerBlock = blockDim.x >> 5;
    int gw = blockIdx.x * wavesPerBlock + (threadIdx.x >> 5);
    int totalWaves = rowTiles * 4;          // 4 column groups of 64
    if (gw >= totalWaves) return;           // wave-uniform exit (EXEC stays full)

    int rowTile = gw >> 2;
    int colGrp  = gw & 3;
    int lane = threadIdx.x & 31;
    int l15  = lane & 15;
    int khalf = (lane >> 4) * 2;            // K sub-offset: 0 or 2

    // A fragment row for this lane (clamped for partial last tile; masked later)
    int rowA = rowTile * 16 + l15;
    int rowL = (rowA < N) ? rowA : (N - 1);
    float invc = 1.0f / fmaxf(cnt[rowL], 1.0f);
    const float* sRow = s  + (long)rowL * CFEAT;
    const float* xRow = xt + (long)rowL * CFEAT;

    int colBase = colGrp * 64;
    const float* wlRow[4];
    const float* wrRow[4];
#pragma unroll
    for (int t = 0; t < 4; ++t) {
        int colB = colBase + t * 16 + l15;  // B fragment: lane = output column
        wlRow[t] = Wl + (long)colB * CFEAT;
        wrRow[t] = Wr + (long)colB * CFEAT;
    }

    v8f acc[4] = {{0,0,0,0,0,0,0,0}, {0,0,0,0,0,0,0,0},
                  {0,0,0,0,0,0,0,0}, {0,0,0,0,0,0,0,0}};

    for (int k = 0; k < CFEAT; k += 4) {
        int ko = k + khalf;
        v2f aL = *(const v2f*)(sRow + ko);
        aL *= invc;                          // fold mean divide into A fragment
        v2f aR = *(const v2f*)(xRow + ko);
#pragma unroll
        for (int t = 0; t < 4; ++t) {
            v2f bL = *(const v2f*)(wlRow[t] + ko);
            v2f bR = *(const v2f*)(wrRow[t] + ko);
            acc[t] = __builtin_amdgcn_wmma_f32_16x16x4_f32(
                false, aL, false, bL, (short)0, acc[t], false, false);
            acc[t] = __builtin_amdgcn_wmma_f32_16x16x4_f32(
                false, aR, false, bR, (short)0, acc[t], false, false);
        }
    }

    // Epilogue: bias + ReLU + store + per-column sum / sum^2 (masking padded rows)
    int rbase = rowTile * 16 + ((lane < 16) ? 0 : 8);
#pragma unroll
    for (int t = 0; t < 4; ++t) {
        int col = colBase + t * 16 + l15;
        float bias = bl[col];
        float sp = 0.0f, qp = 0.0f;
#pragma unroll
        for (int v = 0; v < 8; ++v) {
            float val = fmaxf(acc[t][v] + bias, 0.0f);
            int r = rbase + v;
            bool ok = (r < N);
            float m = ok ? val : 0.0f;
            sp += m;
            qp += m * m;
            if (ok) h[(long)r * CFEAT + col] = val;
        }
        atomicAdd(&colsum[col], sp);
        atomicAdd(&colsq[col], qp);
    }
}

// ---------------------------------------------------------------- axis-0 normalization
__global__ void norm_kernel(const float* __restrict__ hbuf,
                            const float* __restrict__ colsum,
                            const float* __restrict__ colsq,
                            const float* __restrict__ gamma,
                            const float* __restrict__ beta,
                            float* __restrict__ out, int N) {
    long i = (long)blockIdx.x * blockDim.x + threadIdx.x;
    long total = (long)N * CFEAT;
    if (i >= total) return;
    int col = (int)(i & (CFEAT - 1));
    float invN = 1.0f / (float)N;
    float mu  = colsum[col] * invN;
    float var = colsq[col] * invN - mu * mu;
    float rs  = rsqrtf(var + 1e-5f);
    out[i] = gamma[col] * (hbuf[i] - mu) * rs + beta[col];
}

// ---------------------------------------------------------------- driver
extern "C" void kernel_launch(void* const* d_in, const int* in_sizes, int n_in,
                              void* d_out, int out_size, void* d_ws, size_t ws_size,
                              hipStream_t stream) {
    const float* x    = (const float*)d_in[0];
    const int* e0src  = (const int*)d_in[1];
    const int* e0dst  = (const int*)d_in[2];
    const int* e1src  = (const int*)d_in[3];
    const int* e1dst  = (const int*)d_in[4];
    const float* Wl0  = (const float*)d_in[5];
    const float* bl0  = (const float*)d_in[6];
    const float* Wr0  = (const float*)d_in[7];
    const float* g0   = (const float*)d_in[8];
    const float* be0  = (const float*)d_in[9];
    const float* Wl1  = (const float*)d_in[10];
    const float* bl1  = (const float*)d_in[11];
    const float* Wr1  = (const float*)d_in[12];
    const float* g1   = (const float*)d_in[13];
    const float* be1  = (const float*)d_in[14];

    const int N1 = 50000, N2 = 12500;
    const int E0 = in_sizes[1];   // 800000
    const int E1 = in_sizes[3];   // 400000

    // workspace layout (floats)
    float* ws = (float*)d_ws;
    const long NC1 = (long)N1 * CFEAT;      // 12.8M floats
    float* x1    = ws;                      // layer-0 output / layer-1 input
    float* sbuf  = ws + NC1;                // segment sums
    float* hbuf  = ws + 2 * NC1;            // pre-norm activations
    float* cntb  = ws + 3 * NC1;            // counts (<= 50000)
    float* stats = cntb + 50048;            // sum[256], sumsq[256]

    // ---------------- layer 0 ----------------
    zero_kernel<<<4096, 256, 0, stream>>>(sbuf, NC1);
    zero_kernel<<<256, 256, 0, stream>>>(cntb, 50048 + 512);
    scatter_kernel<<<(E0 + 7) / 8, 256, 0, stream>>>(x, e0src, e0dst, sbuf, cntb, E0);
    {
        int rowTiles = (N1 + 15) / 16;      // 3125
        int waves = rowTiles * 4;
        gemm_kernel<<<(waves + 7) / 8, 256, 0, stream>>>(
            sbuf, cntb, x, Wl0, bl0, Wr0, hbuf, stats, stats + 256, N1, rowTiles);
    }
    norm_kernel<<<(int)((NC1 + 255) / 256), 256, 0, stream>>>(
        hbuf, stats, stats + 256, g0, be0, x1, N1);

    // ---------------- layer 1 ----------------
    const long NC2 = (long)N2 * CFEAT;      // 3.2M floats
    zero_kernel<<<2048, 256, 0, stream>>>(sbuf, NC2);
    zero_kernel<<<256, 256, 0, stream>>>(cntb, 50048 + 512);
    scatter_kernel<<<(E1 + 7) / 8, 256, 0, stream>>>(x1, e1src, e1dst, sbuf, cntb, E1);
    {
        int rowTiles = (N2 + 15) / 16;      // 782 (last tile partially masked)
        int waves = rowTiles * 4;
        gemm_kernel<<<(waves + 7) / 8, 256, 0, stream>>>(
            sbuf, cntb, x1, Wl1, bl1, Wr1, hbuf, stats, stats + 256, N2, rowTiles);
    }
    norm_kernel<<<(int)((NC2 + 255) / 256), 256, 0, stream>>>(
        hbuf, stats, stats + 256, g1, be1, (float*)d_out, N2);
}